// CNoiseProjectorBisect_76218489634959
// MI455X (gfx1250) — compile-verified
//
#include <hip/hip_runtime.h>
#include <math.h>

#define AS1 __attribute__((address_space(1)))
#define AS3 __attribute__((address_space(3)))

// Native vector types (HIP's float4 is a class type -> rejected by
// __builtin_nontemporal_store; builtin async-LDS wants int __vector(4) ptrs).
typedef float vf4 __attribute__((ext_vector_type(4)));
typedef int   vi4 __attribute__((vector_size(16)));

// ---- CDNA5 async global->LDS path (guarded so compile never breaks) ----
#if defined(__gfx1250__) && __has_builtin(__builtin_amdgcn_global_load_async_to_lds_b128)
#define USE_ASYNC_LDS 1
#else
#define USE_ASYNC_LDS 0
#endif

#if USE_ASYNC_LDS
__device__ __forceinline__ void wait_asynccnt_le(int n) {
#if __has_builtin(__builtin_amdgcn_s_wait_asynccnt)
  if (n == 0) __builtin_amdgcn_s_wait_asynccnt(0);
  else        __builtin_amdgcn_s_wait_asynccnt(1);
#else
  if (n == 0) asm volatile("s_wait_asynccnt 0x0" ::: "memory");
  else        asm volatile("s_wait_asynccnt 0x1" ::: "memory");
#endif
}
#endif

namespace {

constexpr int BLOCK = 256;
constexpr int VEC   = 4;
constexpr int TILE  = BLOCK * VEC;   // 1024 floats = 4 KB per tile

// Bisection of c(s) = (log(exp(2s)) + log(exp(1.5s)))/8 on [0,80].
// fp32 collapse #1: log(exp(x)) == x (+-1 ulp) until exp overflows at
//   x > 88.722839, past which c -> +inf. So c(mid) = 0.4375*mid with a jump
//   to +inf at mid > 44.3614196 (the s where exp(2s) overflows).
// fp32 collapse #2: the bracket test  c(mid) < ct  is then equivalent to
//   mid < target, target = min(ct/0.4375, 44.3614196)
// computed ONCE per element: the min() covers both ct==inf and finite ct
// above c(44.36142-) = 19.408, where the reference bisection pins to the
// overflow boundary. Loop body is 5 full-rate VALU ops, zero transcendentals;
// only 4 trans ops remain in the whole evaluation.
__device__ __forceinline__ void cnoise_eval(float t, float r[5]) {
  const float EPSf  = 1e-18f;
  const float S_OVF = 44.3614196777f;   // exp(2s) fp32 overflow boundary
  const float INV_C1 = 8.0f / 3.5f;     // 1/0.4375

  float gw = __expf(t);                 // gw_dot == gw
  float ct = 0.25f * __logf(fmaxf(gw, EPSf));
  ct = fmaxf(ct, 0.0f);                 // clip to [c(0), c(T)] = [0, +inf)

  const float target = fminf(ct * INV_C1, S_OVF);

  float lo = 0.0f, hi = 80.0f;
#pragma unroll
  for (int i = 0; i < 32; ++i) {
    float mid  = 0.5f * (lo + hi);
    bool right = mid < target;          // == (c(mid) < ct), increasing branch
    lo = right ? mid : lo;
    hi = right ? hi  : mid;
  }
  float s  = 0.5f * (lo + hi);

  float g  = __expf(2.0f * s);
  float h  = __expf(1.5f * s);
  float g1 = 2.0f * g;
  float h1 = 1.5f * h;
  // keep the faithful divisions: at the overflow-boundary root g may be inf,
  // making c1 (and thus dgdt/dhdt) NaN exactly like the reference.
  float c1 = (g1 / fmaxf(g, EPSf) + h1 / fmaxf(h, EPSf)) * 0.125f;
  float dsdt = 0.25f * (gw / fmaxf(gw, EPSf)) / fmaxf(c1, EPSf);

  r[0] = g;
  r[1] = g1 * dsdt;
  r[2] = h;
  r[3] = h1 * dsdt;
  r[4] = s;
}

__global__ void __launch_bounds__(BLOCK)
cnoise_bisect_kernel(const float* __restrict__ t_in,
                     float* __restrict__ out,
                     int n, int numTiles) {
#if USE_ASYNC_LDS
  __shared__ float smem[2 * TILE];
#endif
  const int tid    = threadIdx.x;
  const int stride = gridDim.x;
  const long nl    = (long)n;

#if USE_ASYNC_LDS
  // Stage this thread's 16B of a tile into LDS via the async DMA engine:
  // hides HBM latency for the next tile behind the VALU work of the current
  // one without holding the in-flight data in VGPRs.
  auto issue = [&](int tileIdx, int buf) {
    long base = (long)tileIdx * TILE + (long)tid * VEC;
    if (base + (VEC - 1) < nl) {
      AS1 vi4* gsrc = (AS1 vi4*)(t_in + base);
      AS3 vi4* ldst = (AS3 vi4*)(&smem[buf * TILE + tid * VEC]);
      __builtin_amdgcn_global_load_async_to_lds_b128(gsrc, ldst, 0, 0);
    }
  };
  if (blockIdx.x < (unsigned)numTiles) issue(blockIdx.x, 0);
#endif

  int buf = 0;
  for (int it = blockIdx.x; it < numTiles; it += stride) {
    const int nextIt = it + stride;
#if USE_ASYNC_LDS
    if (nextIt < numTiles) {
      issue(nextIt, buf ^ 1);      // prefetch next tile into other buffer
      wait_asynccnt_le(1);         // loads retire in order: current tile ready
    } else {
      wait_asynccnt_le(0);
    }
#endif

    const long base = (long)it * TILE + (long)tid * VEC;
    if (base + (VEC - 1) < nl) {
      vf4 tv;
#if USE_ASYNC_LDS
      tv = *(const vf4*)(&smem[buf * TILE + tid * VEC]);
#else
      tv = *(const vf4*)(t_in + base);
#endif
      float r0[5], r1[5], r2[5], r3[5];
      cnoise_eval(tv.x, r0);
      cnoise_eval(tv.y, r1);
      cnoise_eval(tv.z, r2);
      cnoise_eval(tv.w, r3);
      // 5 output streams, write-once (320MB total > 192MB L2): NT stores.
#pragma unroll
      for (int q = 0; q < 5; ++q) {
        vf4 v;
        v.x = r0[q]; v.y = r1[q]; v.z = r2[q]; v.w = r3[q];
        __builtin_nontemporal_store(v, (vf4*)(out + (long)q * nl + base));
      }
    } else if (base < nl) {
      // ragged tail: scalar path
      for (int k = 0; k < VEC; ++k) {
        long i = base + k;
        if (i < nl) {
          float r[5];
          cnoise_eval(t_in[i], r);
#pragma unroll
          for (int q = 0; q < 5; ++q) out[(long)q * nl + i] = r[q];
        }
      }
    }
    buf ^= 1;
  }
}

} // namespace

extern "C" void kernel_launch(void* const* d_in, const int* in_sizes, int n_in,
                              void* d_out, int out_size, void* d_ws, size_t ws_size,
                              hipStream_t stream) {
  (void)n_in; (void)out_size; (void)d_ws; (void)ws_size;
  const float* t = (const float*)d_in[0];
  float* out     = (float*)d_out;
  const int n    = in_sizes[0];

  int numTiles = (n + TILE - 1) / TILE;
  if (numTiles < 1) numTiles = 1;
  int grid = numTiles < 4096 ? numTiles : 4096;   // >=4 tiles/block at N=2^24
                                                  // so the double buffer pipelines
  cnoise_bisect_kernel<<<grid, BLOCK, 0, stream>>>(t, out, n, numTiles);
}